// GCNLayer_566935683469
// MI455X (gfx1250) — compile-verified
//
#include <hip/hip_runtime.h>
#include <hip/hip_bf16.h>

// ---------------------------------------------------------------------------
// GCN layer, restructured:   out = segsum_row( val * (X @ W^T)[col] ) + b
// Phase 1: Y = X @ W^T          -- dense f32 WMMA GEMM (V_WMMA_F32_16X16X4_F32)
// Phase 2: out[i][:] = b[:]     -- bias broadcast init
// Phase 3: out[row[e]] += val[e] * Y[col[e]]   -- f32 atomic scatter (L2-resident)
// ---------------------------------------------------------------------------

#define DDIM 256
#define KC   64
#define LDSS (DDIM + 4)   // padded LDS row stride (floats) to spread banks

typedef float v2f __attribute__((ext_vector_type(2)));
typedef float v8f __attribute__((ext_vector_type(8)));

// ---------------------------------------------------------------------------
// Phase 1: Y[n, o] = sum_k X[n, k] * W[o, k]
// Block = 256 threads = 8 waves. Each wave computes one 16-row M-tile x 256 cols.
// W is staged transposed into LDS in K-chunks of KC.
// ---------------------------------------------------------------------------
__global__ __launch_bounds__(256) void gcn_gemm_xwT(
    const float* __restrict__ X, const float* __restrict__ W,
    float* __restrict__ Y, int n_rows)
{
    __shared__ float ldsB[KC * LDSS];   // ldsB[kk][o] = W[o][kc + kk]

    const int tid  = threadIdx.x;
    const int lane = tid & 31;
    const int wave = tid >> 5;
    const int lrow = lane & 15;   // M index within tile (A) / N col (B, C, D)
    const int lhi  = lane >> 4;   // lane-half select

    const int mtile = blockIdx.x * 8 + wave;
    const int row0  = mtile * 16;
    const bool active = (row0 < n_rows);   // wave-uniform: EXEC all-1s inside

    v8f acc[16] = {};   // 16 N-tiles of 16x16 f32 accumulators

    for (int kc = 0; kc < DDIM; kc += KC) {
        __syncthreads();
        // Cooperative transpose-load of W[:, kc:kc+KC] into LDS (coalesced reads)
        for (int idx = tid; idx < DDIM * KC; idx += 256) {
            int o  = idx >> 6;     // output channel
            int kk = idx & (KC - 1);
            ldsB[kk * LDSS + o] = W[o * DDIM + kc + kk];
        }
        __syncthreads();

        if (active) {
            for (int ks = 0; ks < KC; ks += 4) {
                // A fragment (16x4 f32): lane holds X[row0+lrow][k + 2*lhi + {0,1}]
                const float* ap = X + (size_t)(row0 + lrow) * DDIM + kc + ks + 2 * lhi;
                v2f a;
                a.x = ap[0];
                a.y = ap[1];
                #pragma unroll
                for (int nt = 0; nt < 16; ++nt) {
                    // B fragment (4x16 f32): VGPR j holds Wt[ks + j + 2*lhi][nt*16 + lrow]
                    const float* bp = &ldsB[(ks + 2 * lhi) * LDSS + nt * 16 + lrow];
                    v2f b;
                    b.x = bp[0];
                    b.y = bp[LDSS];
                    acc[nt] = __builtin_amdgcn_wmma_f32_16x16x4_f32(
                        /*neg_a=*/false, a, /*neg_b=*/false, b,
                        /*c_mod=*/(short)0, acc[nt],
                        /*reuse_a=*/false, /*reuse_b=*/false);
                }
            }
        }
    }

    if (active) {
        // C/D layout: VGPR r, lane -> row (r + 8*lhi), col lrow
        #pragma unroll
        for (int nt = 0; nt < 16; ++nt) {
            #pragma unroll
            for (int r = 0; r < 8; ++r) {
                Y[(size_t)(row0 + r + 8 * lhi) * DDIM + nt * 16 + lrow] = acc[nt][r];
            }
        }
    }
}

// ---------------------------------------------------------------------------
// Phase 2: out[i*256 + j] = b[j]
// ---------------------------------------------------------------------------
__global__ __launch_bounds__(256) void gcn_bias_init(
    float* __restrict__ out, const float* __restrict__ b, int total)
{
    int i = blockIdx.x * 256 + threadIdx.x;
    if (i < total) out[i] = b[i & (DDIM - 1)];
}

// ---------------------------------------------------------------------------
// Phase 3: out[row[e]][:] += val[e] * Y[col[e]][:]
// 64 threads per edge, float4 per thread; Y rows are L2-resident (102MB < 192MB L2)
// ---------------------------------------------------------------------------
__global__ __launch_bounds__(256) void gcn_edge_scatter(
    const int* __restrict__ erow, const int* __restrict__ ecol,
    const float* __restrict__ eval, const float* __restrict__ Y,
    float* __restrict__ out, int n_edges)
{
    long long gid = (long long)blockIdx.x * 256 + threadIdx.x;
    int e = (int)(gid >> 6);
    int t = ((int)gid & 63) * 4;
    if (e >= n_edges) return;

    int   r = erow[e];
    int   c = ecol[e];
    float v = eval[e];

    const float4 y = *(const float4*)(Y + (size_t)c * DDIM + t);
    float* dst = out + (size_t)r * DDIM + t;
    unsafeAtomicAdd(dst + 0, v * y.x);
    unsafeAtomicAdd(dst + 1, v * y.y);
    unsafeAtomicAdd(dst + 2, v * y.z);
    unsafeAtomicAdd(dst + 3, v * y.w);
}

// ---------------------------------------------------------------------------
extern "C" void kernel_launch(void* const* d_in, const int* in_sizes, int n_in,
                              void* d_out, int out_size, void* d_ws, size_t ws_size,
                              hipStream_t stream)
{
    const float* X    = (const float*)d_in[0];
    const int*   erow = (const int*)  d_in[1];
    const int*   ecol = (const int*)  d_in[2];
    const float* ev   = (const float*)d_in[3];
    const float* W    = (const float*)d_in[4];
    const float* b    = (const float*)d_in[5];
    float*       out  = (float*)d_out;
    float*       Y    = (float*)d_ws;   // needs N*256*4 = 102.4 MB scratch

    const int n_nodes = in_sizes[0] / DDIM;
    const int n_edges = in_sizes[1];

    // Phase 1: Y = X @ W^T via f32 WMMA
    int mtiles = (n_nodes + 15) / 16;
    int gemm_blocks = (mtiles + 7) / 8;
    gcn_gemm_xwT<<<gemm_blocks, 256, 0, stream>>>(X, W, Y, n_nodes);

    // Phase 2: out = broadcast bias
    int total = n_nodes * DDIM;
    gcn_bias_init<<<(total + 255) / 256, 256, 0, stream>>>(out, b, total);

    // Phase 3: atomic edge scatter
    long long sthreads = (long long)n_edges * 64;
    int sblocks = (int)((sthreads + 255) / 256);
    gcn_edge_scatter<<<sblocks, 256, 0, stream>>>(erow, ecol, ev, Y, out, n_edges);
}